// SE_44126493999479
// MI455X (gfx1250) — compile-verified
//
#include <hip/hip_runtime.h>

typedef __attribute__((ext_vector_type(2))) float v2f;
typedef __attribute__((ext_vector_type(4))) float v4f;
typedef __attribute__((ext_vector_type(8))) float v8f;

#define BATCH 64
#define HGT 28
#define WID 28
#define CH 960
#define CRED 240
#define HW (HGT * WID)          // 784
#define SPLIT 14
#define CHUNK (HW / SPLIT)      // 56

// ---------------- zero the pooled-sum buffer ----------------
__global__ void se_zero(float* __restrict__ s, int n) {
    int i = blockIdx.x * blockDim.x + threadIdx.x;
    if (i < n) s[i] = 0.0f;
}

// ---------------- global average pool ----------------
// grid (BATCH, SPLIT), block = CH threads (1 thread per channel, coalesced)
__global__ void se_pool(const float* __restrict__ x, float* __restrict__ s) {
    int b = blockIdx.x;
    int chunk = blockIdx.y;
    int c = threadIdx.x;  // 0..959
    const float* p = x + ((size_t)b * HW + (size_t)chunk * CHUNK) * CH + c;
    float acc = 0.0f;
#pragma unroll 8
    for (int i = 0; i < CHUNK; ++i) acc += p[(size_t)i * CH];
    atomicAdd(&s[b * CH + c], acc * (1.0f / (float)HW));
}

// ---------------- FC1: e1 = relu(s @ w1)  [64,960]x[960,240] ----------------
// grid = 15 (N tiles), block = 128 (4 waves = 4 M tiles of the 64-row batch)
__global__ void se_fc1(const float* __restrict__ s, const float* __restrict__ w1,
                       float* __restrict__ e1) {
    int lane = threadIdx.x & 31;
    int wave = threadIdx.x >> 5;      // M tile 0..3
    int nt = blockIdx.x;              // N tile 0..14
    int half = lane >> 4;             // 0: K{0,1} / 1: K{2,3}
    int l = lane & 15;
    int row = wave * 16 + l;          // A-matrix row (M)
    int col = nt * 16 + l;            // B/D column (N)

    const float* pa = s + (size_t)row * CH + 2 * half;
    const float* pb = w1 + (size_t)(2 * half) * CRED + col;

    v8f acc = {};
    for (int k = 0; k < CH; k += 4) {
        v2f a, b;
        a.x = pa[0];     a.y = pa[1];       // A: K = 2*half, 2*half+1
        b.x = pb[0];     b.y = pb[CRED];    // B: rows K = 2*half, 2*half+1
        acc = __builtin_amdgcn_wmma_f32_16x16x4_f32(
            false, a, false, b, (short)0, acc, false, false);
        pa += 4;
        pb += (size_t)4 * CRED;
    }
    // D layout: acc[r] = out[wave*16 + r + 8*half][col]
#pragma unroll
    for (int r = 0; r < 8; ++r) {
        float v = acc[r];
        v = v < 0.0f ? 0.0f : v;  // relu
        e1[(size_t)(wave * 16 + r + 8 * half) * CRED + col] = v;
    }
}

// ---------------- FC2: e2 = h_sigmoid(e1 @ w2)  [64,240]x[240,960] ----------------
// grid = 60 (N tiles), block = 128 (4 waves = 4 M tiles)
__global__ void se_fc2(const float* __restrict__ e1, const float* __restrict__ w2,
                       float* __restrict__ e2) {
    int lane = threadIdx.x & 31;
    int wave = threadIdx.x >> 5;
    int nt = blockIdx.x;              // 0..59
    int half = lane >> 4;
    int l = lane & 15;
    int row = wave * 16 + l;
    int col = nt * 16 + l;

    const float* pa = e1 + (size_t)row * CRED + 2 * half;
    const float* pb = w2 + (size_t)(2 * half) * CH + col;

    v8f acc = {};
    for (int k = 0; k < CRED; k += 4) {
        v2f a, b;
        a.x = pa[0];   a.y = pa[1];
        b.x = pb[0];   b.y = pb[CH];
        acc = __builtin_amdgcn_wmma_f32_16x16x4_f32(
            false, a, false, b, (short)0, acc, false, false);
        pa += 4;
        pb += (size_t)4 * CH;
    }
#pragma unroll
    for (int r = 0; r < 8; ++r) {
        float v = acc[r];
        // h_sigmoid: clamp(v+3, 0, 6) / 6
        v = fminf(fmaxf(v + 3.0f, 0.0f), 6.0f) * (1.0f / 6.0f);
        e2[(size_t)(wave * 16 + r + 8 * half) * CH + col] = v;
    }
}

// ---------------- broadcast scale: out = x * e2[b, c] ----------------
// grid (735, BATCH), block 256; one float4 per thread; 735*256 = HW*CH/4 exactly
__global__ void se_scale(const float* __restrict__ x, const float* __restrict__ e2,
                         float* __restrict__ out) {
    int b = blockIdx.y;
    size_t i4 = (size_t)blockIdx.x * blockDim.x + threadIdx.x;
    const size_t n4 = (size_t)HW * CH / 4;  // 188160
    if (i4 >= n4) return;
    size_t off = (size_t)b * HW * CH + i4 * 4;
    int c = (int)((i4 * 4) % CH);
    v4f xin = *(const v4f*)(x + off);
    v4f g = *(const v4f*)(e2 + (size_t)b * CH + c);
    v4f o = xin * g;
    *(v4f*)(out + off) = o;
}

extern "C" void kernel_launch(void* const* d_in, const int* in_sizes, int n_in,
                              void* d_out, int out_size, void* d_ws, size_t ws_size,
                              hipStream_t stream) {
    const float* x  = (const float*)d_in[0];   // [64,28,28,960]
    const float* w1 = (const float*)d_in[1];   // [960,240]
    const float* w2 = (const float*)d_in[2];   // [240,960]
    float* out = (float*)d_out;

    float* s  = (float*)d_ws;            // [64,960]
    float* e1 = s + BATCH * CH;          // [64,240]
    float* e2 = e1 + BATCH * CRED;       // [64,960]

    se_zero<<<(BATCH * CH + 255) / 256, 256, 0, stream>>>(s, BATCH * CH);
    se_pool<<<dim3(BATCH, SPLIT), CH, 0, stream>>>(x, s);
    se_fc1<<<15, 128, 0, stream>>>(s, w1, e1);
    se_fc2<<<60, 128, 0, stream>>>(e1, w2, e2);
    se_scale<<<dim3((HW * CH / 4 + 255) / 256, BATCH), 256, 0, stream>>>(x, e2, out);
}